// Sphere_80582176408183
// MI455X (gfx1250) — compile-verified
//
#include <hip/hip_runtime.h>
#include <hip/hip_bf16.h>

typedef __attribute__((ext_vector_type(2))) float v2f;
typedef __attribute__((ext_vector_type(8))) float v8f;

namespace {
constexpr int kN = 16384;   // points per batch
constexpr int kG = 257;     // sphere-grid centers
constexpr int kGT = 17;     // center tiles of 16 (padded to 272)
constexpr int kK = 32;      // neighbors per center
constexpr int kWaves = 8;   // waves per block
constexpr int kBlock = kWaves * 32;
}

// order-preserving float -> u32 (total order, matches float <)
__device__ __forceinline__ unsigned ford(float f) {
  unsigned u = __float_as_uint(f);
  return u ^ ((unsigned)((int)u >> 31) | 0x80000000u);
}

__device__ __forceinline__ unsigned long long shflx64(unsigned long long v, int mask) {
  int lo = __shfl_xor((int)(unsigned)v, mask, 32);
  int hi = __shfl_xor((int)(unsigned)(v >> 32), mask, 32);
  return ((unsigned long long)(unsigned)hi << 32) | (unsigned)lo;
}

__global__ __launch_bounds__(kBlock) void Sphere_knn_kernel(
    const float* __restrict__ xyz, const float* __restrict__ labels,
    const float* __restrict__ inside, float* __restrict__ out) {
  extern __shared__ char smraw[];
  float4* sm4 = (float4*)smraw;          // [kN + 16] = (x, y, z, |p|^2)
  const int b = blockIdx.x;
  const int nB = gridDim.x;
  const int tid = threadIdx.x;
  const int lane = tid & 31;
  const int wave = tid >> 5;

  const size_t OFF_CEN = (size_t)nB * kG * kK * 3;          // after neighborhood
  const size_t OFF_LAB = OFF_CEN + (size_t)nB * kG * 3;     // after center
  const size_t OFF_IDX = OFF_LAB + (size_t)nB * kG * kK;    // after labels
  const size_t OFF_INS = OFF_IDX + (size_t)nB * kG * kK;    // after idx

  // ---- Stage this batch's xyz into LDS, precomputing |p|^2 once ----
  const float* xb = xyz + (size_t)b * kN * 3;
  for (int n = tid; n < kN; n += kBlock) {
    float x = xb[3 * n + 0];
    float y = xb[3 * n + 1];
    float z = xb[3 * n + 2];
    sm4[n] = make_float4(x, y, z, x * x + y * y + z * z);
  }
  if (tid < 16) sm4[kN + tid] = make_float4(0.f, 0.f, 0.f, 0.f);  // prefetch pad
  __syncthreads();

  const int ct = blockIdx.y * kWaves + wave;   // center tile index
  if (ct >= kGT) return;

  const int hi = lane >> 4;   // 0 = lower half (K=0,1), 1 = upper half (K=2,3)
  const int m = lane & 15;    // A row / B,C,D column within tile
  const int myrank = ct * 16 + m;

  // ---- Regenerate this lane's center (rank myrank) from the 0.25 lattice ----
  float cx = 0.f, cy = 0.f, cz = 0.f;
  {
    int cnt = 0;
#pragma unroll 1
    for (int i = 0; i < 9; ++i) {
      float vi = -1.0f + 0.25f * i;
#pragma unroll 1
      for (int j = 0; j < 9; ++j) {
        float vj = -1.0f + 0.25f * j;
#pragma unroll 1
        for (int k = 0; k < 9; ++k) {
          float vk = -1.0f + 0.25f * k;
          float n2 = vi * vi + vj * vj + vk * vk;
          if (n2 <= 1.000002f) {              // |g| <= 1 + 1e-6 (lattice sums exact)
            if (cnt == myrank) { cx = vi; cy = vj; cz = vk; }
            ++cnt;
          }
        }
      }
    }
  }

  // center output (one writer per (b,g))
  if (hi == 0 && myrank < kG) {
    float* oc = out + OFF_CEN + ((size_t)b * kG + myrank) * 3;
    oc[0] = cx; oc[1] = cy; oc[2] = cz;
  }

  const float c2 = cx * cx + cy * cy + cz * cz;
  // A(16x4): row m = (-2cx, -2cy, -2cz, 1); lane<16 holds K=0,1; lane>=16 K=2,3
  v2f Am;
  Am[0] = hi ? (-2.0f * cz) : (-2.0f * cx);
  Am[1] = hi ? 1.0f : (-2.0f * cy);
  // C(16x16): C[m][n] = |c_m|^2 ; VGPR r holds row r (lo half) / r+8 (hi half)
  v8f Cm;
#pragma unroll
  for (int r = 0; r < 8; ++r) Cm[r] = __shfl(c2, r + hi * 8, 32);

  unsigned long long prev_k[8], curr_k[8];
#pragma unroll
  for (int r = 0; r < 8; ++r) prev_k[r] = 0ull;

  for (int p = 0; p < kK; ++p) {
#pragma unroll
    for (int r = 0; r < 8; ++r) curr_k[r] = ~0ull;

    // software-pipelined tile loop: next tile's ds_load_b128 is in flight
    // while the current tile's WMMA + key-selection VALU work executes.
    float4 pt = sm4[m];
#pragma unroll 2
    for (int t = 0; t < kN; t += 16) {
      const int n = t + m;
      float4 nxt = sm4[n + 16];                 // prefetch (pad tile on last iter)
      // B(4x16): col n = (x, y, z, |p|^2); lane<16 holds K=0,1; lane>=16 K=2,3
      v2f Bm;
      Bm[0] = hi ? pt.z : pt.x;
      Bm[1] = hi ? pt.w : pt.y;
      // D = A*B + C = squared distance tile (16 centers x 16 points), fp32
      v8f D = __builtin_amdgcn_wmma_f32_16x16x4_f32(
          false, Am, false, Bm, (short)0, Cm, false, false);
      const unsigned long long lo = (unsigned)n;
#pragma unroll
      for (int r = 0; r < 8; ++r) {
        unsigned long long key = ((unsigned long long)ford(D[r]) << 32) | lo;
        bool upd = (key > prev_k[r]) & (key < curr_k[r]);   // branch-free
        curr_k[r] = upd ? key : curr_k[r];
      }
      pt = nxt;
    }

    // reduce key-min across each 16-lane half (lexicographic (d, idx) min), emit
#pragma unroll
    for (int r = 0; r < 8; ++r) {
      unsigned long long k = curr_k[r];
#pragma unroll
      for (int s = 1; s < 16; s <<= 1) {
        unsigned long long o = shflx64(k, s);
        k = (o < k) ? o : k;
      }
      prev_k[r] = k;

      const int g = ct * 16 + r + hi * 8;
      float rcx = __shfl(cx, r + hi * 8, 32);   // center coords of this row
      float rcy = __shfl(cy, r + hi * 8, 32);
      float rcz = __shfl(cz, r + hi * 8, 32);
      if (m == 0 && g < kG) {                   // lane 0 -> rows 0..7, lane 16 -> 8..15
        const int i = (int)(unsigned)(k & 0xffffffffull);
        float4 pi4 = sm4[i];
        const size_t base = ((size_t)b * kG + g) * kK + p;
        float* on = out + base * 3;             // neighborhood at offset 0
        on[0] = pi4.x - rcx;
        on[1] = pi4.y - rcy;
        on[2] = pi4.z - rcz;
        out[OFF_LAB + base] = labels[(size_t)b * kN + i];
        out[OFF_IDX + base] = (float)i;
        out[OFF_INS + base] = inside[(size_t)b * kN + i];
      }
    }
  }
}

extern "C" void kernel_launch(void* const* d_in, const int* in_sizes, int n_in,
                              void* d_out, int out_size, void* d_ws, size_t ws_size,
                              hipStream_t stream) {
  const float* xyz    = (const float*)d_in[0];
  const float* labels = (const float*)d_in[1];
  const float* inside = (const float*)d_in[2];
  float* out = (float*)d_out;
  const int B = in_sizes[0] / (kN * 3);
  dim3 grid(B, (kGT + kWaves - 1) / kWaves);
  const size_t shmem = (size_t)(kN + 16) * sizeof(float4);  // ~256 KB CDNA5 WGP LDS
  hipLaunchKernelGGL(Sphere_knn_kernel, grid, dim3(kBlock), shmem, stream,
                     xyz, labels, inside, out);
}